// XCAttention_32804960207148
// MI455X (gfx1250) — compile-verified
//
#include <hip/hip_runtime.h>
#include <hip/hip_bf16.h>
#include <math.h>

// Problem constants (match reference)
#define BATCH 2
#define CC    64          // channels
#define C3    192         // 3*C
#define HH    512
#define WWI   512
#define HWsz  (HH * WWI)  // 262144
#define HEADS 8
#define WIN   8

typedef __attribute__((ext_vector_type(2))) float v2f;
typedef __attribute__((ext_vector_type(8))) float v8f;
typedef __attribute__((ext_vector_type(4))) int   v4i;
typedef __attribute__((ext_vector_type(8))) int   v8i;

// D = A(16x4, f32) * B(4x16, f32) + C   -- V_WMMA_F32_16X16X4_F32
__device__ __forceinline__ v8f wmma_f32_16x16x4(v2f a, v2f b, v8f c) {
    return __builtin_amdgcn_wmma_f32_16x16x4_f32(
        /*neg_a=*/false, a, /*neg_b=*/false, b,
        /*c_mod=*/(short)0, c, /*reuse_a=*/false, /*reuse_b=*/false);
}

// ---------------------------------------------------------------------------
// TDM: gather a 3D tile (8 channels x 8 rows x 8 contiguous f32) from a
// [C,H,W] f32 tensor into a packed 8x64 LDS tile with ONE instruction.
// Descriptor per CDNA5 ISA:
//   g0: count=1 | lds_addr | global_addr(57b) | type=2
//   g1: data_size=4B, tensor_dim0=W, tensor_dim1=H, tile=8x8x8,
//       dim0_stride=W, dim1_stride=HW
//   g2: tensor_dim2=8, dim2_stride=HW ; g3: unused (zeros)
// Tracked by TENSORcnt; EXEC ignored.
// ---------------------------------------------------------------------------
__device__ __forceinline__ void tdm_load_3d_tile(const float* gptr,
                                                 unsigned lds_off) {
    const unsigned long long ga = (unsigned long long)(uintptr_t)gptr;
    const v4i g0 = { 1,                                  // count=1, user desc
                     (int)lds_off,                       // lds_addr (bytes)
                     (int)(unsigned)(ga & 0xFFFFFFFFu),  // global_addr[31:0]
                     (int)(((unsigned)(ga >> 32) & 0x01FFFFFFu) | 0x80000000u) };
    const v8i g1 = { 0x00020000,            // data_size=2 (4 bytes), mask=0
                     (int)(512u << 16),     // tensor_dim0[15:0]=512 in [31:16]
                     (int)(512u << 16),     // dim0 hi=0 | tensor_dim1 lo=512
                     (int)(8u << 16),       // dim1 hi=0 | tile_dim0=8
                     (int)((8u << 16) | 8u),// tile_dim1=8 | tile_dim2=8
                     512,                   // tensor_dim0_stride lo = W
                     0,                     // dim0_stride hi | dim1_stride lo
                     4 };                   // tensor_dim1_stride[47:16] = HW>>16
    const v4i g2 = { 8,                     // tensor_dim2 = 8
                     0,                     // tensor_dim3 = 0
                     0x40000,               // tensor_dim2_stride lo = HW
                     0 };                   // stride hi | tile_dim3=0
    const v4i g3 = { 0, 0, 0, 0 };
    asm volatile("tensor_load_to_lds %0, %1, %2, %3"
                 :: "s"(g0), "s"(g1), "s"(g2), "s"(g3)
                 : "memory");
}

// ---------------------------------------------------------------------------
// Kernel 1 & 4: 1x1 conv as GEMM.  out[b, o, p] = sum_c w[o, c] * in[b, c, p]
// IC fixed at 64 (K=64 -> 16 WMMA K-steps). One wave computes a 16x16 tile.
// ---------------------------------------------------------------------------
__global__ __launch_bounds__(256)
void conv1x1_wmma_kernel(const float* __restrict__ in,
                         const float* __restrict__ w,
                         float* __restrict__ out,
                         int OC)
{
    const int lane = threadIdx.x & 31;
    const long wid  = (long)blockIdx.x * 8 + (threadIdx.x >> 5);
    const int  MT   = OC >> 4;                 // tiles along output channels
    const int  mt   = (int)(wid % MT);
    const long pg   = wid / MT;                // 16-pixel group over B*HW
    const long pix0 = pg * 16;
    const int  b    = (int)(pix0 / HWsz);
    const long p0   = pix0 - (long)b * HWsz;

    const float* xb = in + (long)b * CC * HWsz;
    const int mrow = mt * 16 + (lane & 15);    // A row (output channel)
    const int nn   = lane & 15;                // B col (pixel)
    const int kkb  = (lane >> 4) * 2;          // K pair within 4-chunk

    v8f acc = {};
    #pragma unroll
    for (int k0 = 0; k0 < CC; k0 += 4) {
        v2f a, bv;
        a.x  = w[mrow * CC + k0 + kkb];
        a.y  = w[mrow * CC + k0 + kkb + 1];
        bv.x = xb[(long)(k0 + kkb)     * HWsz + p0 + nn];
        bv.y = xb[(long)(k0 + kkb + 1) * HWsz + p0 + nn];
        acc = wmma_f32_16x16x4(a, bv, acc);
    }

    float* ob = out + (long)b * OC * HWsz;
    #pragma unroll
    for (int i = 0; i < 8; ++i) {
        const int m = mt * 16 + i + ((lane >> 4) << 3);  // row per D layout
        ob[(long)m * HWsz + p0 + nn] = acc[i];
    }
}

// ---------------------------------------------------------------------------
// Kernel 2: depthwise 3x3, zero padding, 192 channels. 4 pixels per thread.
// ---------------------------------------------------------------------------
__global__ __launch_bounds__(256)
void dwconv3x3_kernel(const float* __restrict__ in,
                      const float* __restrict__ w,
                      float* __restrict__ out)
{
    const long tid = (long)blockIdx.x * blockDim.x + threadIdx.x;
    const int  x4  = (int)(tid % (WWI / 4));
    long t = tid / (WWI / 4);
    const int y  = (int)(t % HH); t /= HH;
    const int ch = (int)(t % C3);
    const int b  = (int)(t / C3);

    const float* ip = in + (long)(b * C3 + ch) * HWsz;
    const float* wp = w + ch * 9;
    float wv[9];
    #pragma unroll
    for (int i = 0; i < 9; ++i) wv[i] = wp[i];

    const int x0 = x4 * 4;
    float o[4] = {0.f, 0.f, 0.f, 0.f};
    #pragma unroll
    for (int dy = -1; dy <= 1; ++dy) {
        const int yy = y + dy;
        if (yy < 0 || yy >= HH) continue;
        const float* row = ip + (long)yy * WWI;
        float r[6];
        #pragma unroll
        for (int j = 0; j < 6; ++j) {
            const int xx = x0 - 1 + j;
            r[j] = (xx >= 0 && xx < WWI) ? row[xx] : 0.f;
        }
        #pragma unroll
        for (int p = 0; p < 4; ++p)
            #pragma unroll
            for (int dx = 0; dx < 3; ++dx)
                o[p] = fmaf(r[p + dx], wv[(dy + 1) * 3 + dx], o[p]);
    }
    float4 v4 = make_float4(o[0], o[1], o[2], o[3]);
    *reinterpret_cast<float4*>(out + (long)(b * C3 + ch) * HWsz +
                               (long)y * WWI + x0) = v4;
}

// ---------------------------------------------------------------------------
// Kernel 3: windowed channel-covariance attention.
// One block per 8x8 window, one wave per head.
//   q,k,v : 8x64 tiles pulled into LDS by the Tensor Data Mover (1 op each)
//   attn  = l2norm(q) . l2norm(k)^T   (8x8)   via WMMA f32 16x16x4
//   softmax over last dim (temperature-scaled)
//   out   = attn . v                 (8x64)   via WMMA f32 16x16x4
// Writes directly in un-windowed [B,C,H,W] layout.
// ---------------------------------------------------------------------------
__global__ __launch_bounds__(256)
void xca_window_attn_kernel(const float* __restrict__ qkv,  // [B,3C,H,W]
                            const float* __restrict__ temperature, // [HEADS]
                            float* __restrict__ out)        // [B,C,H,W]
{
    __shared__ float q_s[HEADS][512];
    __shared__ float k_s[HEADS][512];
    __shared__ float v_s[HEADS][512];
    __shared__ float at_s[HEADS][64];
    __shared__ float qsc[HEADS][8];
    __shared__ float ksc[HEADS][8];

    const int lane = threadIdx.x & 31;
    const int h    = threadIdx.x >> 5;            // head = wave id in block
    const int n    = blockIdx.x;                  // window id 0..8191
    const int wwb  = n & 63;
    const int whb  = (n >> 6) & 63;
    const int b    = n >> 12;

    const float* base = qkv + (long)b * C3 * HWsz;

    // TDM: one 3D-tile DMA per matrix (8 chan x 8 rows x 8 f32 -> packed 8x64)
    const long toff = (long)(h * 8) * HWsz + (long)(whb * WIN) * WWI + wwb * WIN;
    tdm_load_3d_tile(base + toff,
                     (unsigned)(uintptr_t)&q_s[h][0]);
    tdm_load_3d_tile(base + (long)CC * HWsz + toff,
                     (unsigned)(uintptr_t)&k_s[h][0]);
    tdm_load_3d_tile(base + (long)(2 * CC) * HWsz + toff,
                     (unsigned)(uintptr_t)&v_s[h][0]);
    __builtin_amdgcn_s_wait_tensorcnt(0);   // this wave's tiles are in LDS
    __syncthreads();

    // L2-norm scales: lanes 0-7 -> q rows, lanes 8-15 -> k rows
    if (lane < 16) {
        const float* src = (lane < 8) ? q_s[h] : k_s[h];
        const int r = lane & 7;
        float s = 0.f;
        #pragma unroll 8
        for (int j = 0; j < 64; ++j) { const float v = src[r * 64 + j]; s = fmaf(v, v, s); }
        const float sc = 1.f / fmaxf(sqrtf(s), 1e-12f);
        if (lane < 8) qsc[h][r] = sc; else ksc[h][r] = sc;
    }
    __syncthreads();

    const int m   = lane & 15;          // A row / B col index
    const int kkb = (lane >> 4) * 2;    // K pair
    const float sA = (m < 8) ? qsc[h][m] : 0.f;
    const float sB = (m < 8) ? ksc[h][m] : 0.f;

    // attn = qn * kn^T  (rows/cols >= 8 padded with zeros)
    v8f acc = {};
    #pragma unroll
    for (int k0 = 0; k0 < 64; k0 += 4) {
        v2f a, bv;
        a.x  = (m < 8) ? q_s[h][m * 64 + k0 + kkb]     * sA : 0.f;
        a.y  = (m < 8) ? q_s[h][m * 64 + k0 + kkb + 1] * sA : 0.f;
        bv.x = (m < 8) ? k_s[h][m * 64 + k0 + kkb]     * sB : 0.f;
        bv.y = (m < 8) ? k_s[h][m * 64 + k0 + kkb + 1] * sB : 0.f;
        acc = wmma_f32_16x16x4(a, bv, acc);
    }

    // temperature-scaled softmax over the 8 valid columns (lanes 0-7 group)
    const float temp = temperature[h];
    float prob[8];
    #pragma unroll
    for (int i = 0; i < 8; ++i) {
        float v = acc[i] * temp;
        float mx = v;
        mx = fmaxf(mx, __shfl_xor(mx, 1, 8));
        mx = fmaxf(mx, __shfl_xor(mx, 2, 8));
        mx = fmaxf(mx, __shfl_xor(mx, 4, 8));
        const float e = __expf(v - mx);
        float s = e;
        s += __shfl_xor(s, 1, 8);
        s += __shfl_xor(s, 2, 8);
        s += __shfl_xor(s, 4, 8);
        prob[i] = e / s;
    }
    if (lane < 8) {                       // rows 0..7 live in lanes 0..7 of VGPR i
        #pragma unroll
        for (int i = 0; i < 8; ++i) at_s[h][i * 8 + lane] = prob[i];
    }
    __syncthreads();

    // out = attn(8x8) * v(8x64): K=8 (2 WMMA steps), 4 tiles of 16 columns
    #pragma unroll
    for (int nt = 0; nt < 4; ++nt) {
        v8f oacc = {};
        #pragma unroll
        for (int k0 = 0; k0 < 8; k0 += 4) {
            v2f a, bv;
            a.x  = (m < 8) ? at_s[h][m * 8 + k0 + kkb]     : 0.f;
            a.y  = (m < 8) ? at_s[h][m * 8 + k0 + kkb + 1] : 0.f;
            bv.x = v_s[h][(k0 + kkb)     * 64 + nt * 16 + m];
            bv.y = v_s[h][(k0 + kkb + 1) * 64 + nt * 16 + m];
            oacc = wmma_f32_16x16x4(a, bv, oacc);
        }
        if (lane < 16) {                  // valid rows 0..7 (lanes 0-15, VGPR 0-7)
            const int wc = nt * 16 + lane;
            const int yy = whb * WIN + (wc >> 3);
            const int xx = wwb * WIN + (wc & 7);
            #pragma unroll
            for (int i = 0; i < 8; ++i) {
                const int ch = h * 8 + i;
                out[(long)b * CC * HWsz + (long)ch * HWsz + (long)yy * WWI + xx] = oacc[i];
            }
        }
    }
}

// ---------------------------------------------------------------------------
extern "C" void kernel_launch(void* const* d_in, const int* in_sizes, int n_in,
                              void* d_out, int out_size, void* d_ws, size_t ws_size,
                              hipStream_t stream) {
    const float* x      = (const float*)d_in[0];   // [B,C,H,W]
    const float* qkv_w  = (const float*)d_in[1];   // [3C,C]
    const float* dw_w   = (const float*)d_in[2];   // [3C,1,3,3]
    const float* proj_w = (const float*)d_in[3];   // [C,C]
    const float* temp   = (const float*)d_in[4];   // [HEADS]
    float* out = (float*)d_out;

    // Workspace layout (reuse buf0 for attention output after qkv is dead):
    //   buf0: qkv (pre-dw)   [B,3C,H,W]  -> later reused as attn output [B,C,H,W]
    //   buf1: qkv_dw         [B,3C,H,W]
    float* qkv    = (float*)d_ws;
    float* qkv_dw = qkv + (size_t)BATCH * C3 * HWsz;
    float* att_o  = qkv;   // reuse: kernel 3 only reads qkv_dw

    // 1) qkv = x * qkv_w^T  (GEMM M=192,K=64,N=B*HW) via f32 WMMA
    {
        const long waves = (long)(C3 / 16) * ((long)BATCH * HWsz / 16); // 393216
        conv1x1_wmma_kernel<<<(unsigned)(waves / 8), 256, 0, stream>>>(
            x, qkv_w, qkv, C3);
    }
    // 2) depthwise 3x3
    {
        const long threads = (long)BATCH * C3 * HH * (WWI / 4);
        dwconv3x3_kernel<<<(unsigned)(threads / 256), 256, 0, stream>>>(
            qkv, dw_w, qkv_dw);
    }
    // 3) windowed channel attention (8192 windows, 8 heads/block, TDM staging)
    xca_window_attn_kernel<<<BATCH * 64 * 64, 256, 0, stream>>>(
        qkv_dw, temp, att_o);
    // 4) out = att_o * proj_w^T  (GEMM M=64,K=64,N=B*HW)
    {
        const long waves = (long)(CC / 16) * ((long)BATCH * HWsz / 16); // 131072
        conv1x1_wmma_kernel<<<(unsigned)(waves / 8), 256, 0, stream>>>(
            att_o, proj_w, out, CC);
    }
}